// LinearDynamics_48009144435190
// MI455X (gfx1250) — compile-verified
//
#include <hip/hip_runtime.h>

// x_{t+1} = x_t + dt*(x_t @ A + u_t @ B), batch=1024, T=256, d_x=128, d_u=64.
// Bandwidth-bound in theory (~202 MB vs 12.9 GFLOP), latency-bound in practice
// (256 barrier-separated steps) -> keep f32, V_WMMA_F32_16X16X4_F32, and focus
// on pipelining LDS loads within each step.
// v3: fold dt into weights (Ew = I + dt*A, dt*B) so the WMMA accumulator IS
// x_{t+1}; amdgpu_waves_per_eu(2) + sched_barrier to keep all 48 A-operand
// chunks register-staged with pipelined (partial) dscnt waits.

typedef float v2f __attribute__((ext_vector_type(2)));
typedef float v4f __attribute__((ext_vector_type(4)));
typedef float v8f __attribute__((ext_vector_type(8)));

#define T_STEPS 256
#define D_X 128
#define D_U 64
#define DT_C 0.1f
#define XS 132   // x-tile LDS row stride (132 % 64 = 4 -> conflict-free b64 A-op loads)
#define US 68    // u-tile LDS row stride (68  % 64 = 4 -> conflict-free)

__global__ __launch_bounds__(256)
__attribute__((amdgpu_waves_per_eu(2)))
void lindyn_wmma(
    const float* __restrict__ x0,   // [batch, 128]
    const float* __restrict__ u,    // [batch, 256, 64]
    const float* __restrict__ A,    // [128, 128]
    const float* __restrict__ B,    // [64, 128]
    float* __restrict__ out)        // [batch, 257, 128]
{
    __shared__ float xbuf[2][16 * XS];
    __shared__ float ubuf[2][16 * US];

    const int tid  = threadIdx.x;
    const int wv   = tid >> 5;        // wave id 0..7 -> output col tile
    const int lane = tid & 31;
    const int half = lane >> 4;       // 0/1: which 16-lane half
    const int l16  = lane & 15;
    const int rowB = blockIdx.x << 4; // 16 batch rows per workgroup
    const int colj = (wv << 4) + l16; // this wave's output column for this lane

    // ---- Preload folded weights in WMMA B-operand layout ----
    // Ew = I + dt*A and dt*B, so one WMMA pass produces x_{t+1} directly.
    // B-operand (KxN=4x16): lane holds N=l16, VGPR v holds K = 4c + 2*half + v.
    v2f Aw[32];
    v2f Bw[16];
#pragma unroll
    for (int c = 0; c < 32; ++c) {
        const int k0 = 4 * c + 2 * half;
        Aw[c].x = DT_C * A[(k0 + 0) * D_X + colj] + ((k0 + 0) == colj ? 1.0f : 0.0f);
        Aw[c].y = DT_C * A[(k0 + 1) * D_X + colj] + ((k0 + 1) == colj ? 1.0f : 0.0f);
    }
#pragma unroll
    for (int c = 0; c < 16; ++c) {
        const int k0 = 4 * c + 2 * half;
        Bw[c].x = DT_C * B[(k0 + 0) * D_X + colj];
        Bw[c].y = DT_C * B[(k0 + 1) * D_X + colj];
    }

    // ---- Cooperative load of x_0 -> LDS buf0, and emit out[:, t=0, :] ----
    {
        const int row = tid >> 4;
        const int c0  = (tid & 15) * 8;
        v4f a = *(const v4f*)&x0[(rowB + row) * D_X + c0];
        v4f b = *(const v4f*)&x0[(rowB + row) * D_X + c0 + 4];
        *(v4f*)&xbuf[0][row * XS + c0]     = a;
        *(v4f*)&xbuf[0][row * XS + c0 + 4] = b;
        float* o = &out[(size_t)(rowB + row) * (T_STEPS + 1) * D_X + c0];
        *(v4f*)&o[0] = a;
        *(v4f*)&o[4] = b;
    }
    // ---- Cooperative load of u_0 -> LDS buf0 ----
    const int urow = tid >> 4;
    const int uc0  = (tid & 15) * 4;
    {
        v4f a = *(const v4f*)&u[((size_t)(rowB + urow) * T_STEPS) * D_U + uc0];
        *(v4f*)&ubuf[0][urow * US + uc0] = a;
    }
    __syncthreads();

    for (int t = 0; t < T_STEPS; ++t) {
        const int cur = t & 1;
        const int nxt = cur ^ 1;

        // Prefetch u_{t+1} from global while WMMA work proceeds.
        v4f un = {};
        if (t + 1 < T_STEPS)
            un = *(const v4f*)&u[((size_t)(rowB + urow) * T_STEPS + (t + 1)) * D_U + uc0];

        // Stage ALL A-operand chunks into registers, then fence the scheduler
        // so every ds_load issues before the first WMMA (pipelined dscnt waits).
        v2f xa[32];
        v2f ua[16];
        const float* xb = &xbuf[cur][l16 * XS + 2 * half];
#pragma unroll
        for (int c = 0; c < 32; ++c)
            xa[c] = *(const v2f*)&xb[4 * c];    // A-operand: M=l16, K=4c+2*half+v
        const float* ub = &ubuf[cur][l16 * US + 2 * half];
#pragma unroll
        for (int c = 0; c < 16; ++c)
            ua[c] = *(const v2f*)&ub[4 * c];
        __builtin_amdgcn_sched_barrier(0);

        // x_{t+1} = x_t @ (I + dt*A) + u_t @ (dt*B): two independent
        // accumulator chains for WMMA ILP; their sum is the next state.
        v8f acc0 = {};
        v8f acc1 = {};
#pragma unroll
        for (int c = 0; c < 32; c += 2) {
            acc0 = __builtin_amdgcn_wmma_f32_16x16x4_f32(
                false, xa[c + 0], false, Aw[c + 0], (short)0, acc0, false, false);
            acc1 = __builtin_amdgcn_wmma_f32_16x16x4_f32(
                false, xa[c + 1], false, Aw[c + 1], (short)0, acc1, false, false);
        }
#pragma unroll
        for (int c = 0; c < 16; c += 2) {
            acc0 = __builtin_amdgcn_wmma_f32_16x16x4_f32(
                false, ua[c + 0], false, Bw[c + 0], (short)0, acc0, false, false);
            acc1 = __builtin_amdgcn_wmma_f32_16x16x4_f32(
                false, ua[c + 1], false, Bw[c + 1], (short)0, acc1, false, false);
        }

        // Stream x_{t+1} to global and stage it (C/D layout: VGPR r ->
        // M = r + 8*half, N = l16) for the next step's A-operand reads.
#pragma unroll
        for (int r = 0; r < 8; ++r) {
            const float xn = acc0[r] + acc1[r];
            out[((size_t)(rowB + r + 8 * half) * (T_STEPS + 1) + (t + 1)) * D_X + colj] = xn;
            xbuf[nxt][(r + 8 * half) * XS + colj] = xn;
        }
        *(v4f*)&ubuf[nxt][urow * US + uc0] = un;

        __syncthreads();
    }
}

extern "C" void kernel_launch(void* const* d_in, const int* in_sizes, int n_in,
                              void* d_out, int out_size, void* d_ws, size_t ws_size,
                              hipStream_t stream) {
    (void)n_in; (void)out_size; (void)d_ws; (void)ws_size;
    const float* x0 = (const float*)d_in[0];   // initial_state [1024,128]
    const float* u  = (const float*)d_in[1];   // u_traj [1024,256,64]
    const float* A  = (const float*)d_in[2];   // A [128,128]
    const float* B  = (const float*)d_in[3];   // Bmat [64,128]
    float* out = (float*)d_out;                // [1024,257,128]

    const int batch = in_sizes[0] / D_X;       // 1024
    dim3 grid(batch / 16);                     // 64 workgroups
    dim3 block(256);                           // 8 waves: one 16x16 N-tile each
    lindyn_wmma<<<grid, block, 0, stream>>>(x0, u, A, B, out);
}